// GatedTinyMambaLayer_78262894068472
// MI455X (gfx1250) — compile-verified
//
#include <hip/hip_runtime.h>

typedef float v2f __attribute__((ext_vector_type(2)));
typedef float v8f __attribute__((ext_vector_type(8)));
typedef int   v4i __attribute__((ext_vector_type(4)));

#define B_ 64
#define T_ 8192
#define D_ 32
#define S_ 16
#define NTILES (B_ * (T_ / 16))   // 32768
#define BASE_DECAY 0.05f

// padded LDS x-tile row stride (floats): 32 data + 4 pad -> 144B rows,
// makes A-fragment ds_load_b64 reads bank-conflict free.
#define XROW 36

#if __has_builtin(__builtin_amdgcn_global_load_async_to_lds_b128)
#define USE_ASYNC 1
#else
#define USE_ASYNC 0
#endif

typedef __attribute__((address_space(1))) v4i gv4i;   // global int4*
typedef __attribute__((address_space(3))) v4i lv4i;   // LDS int4*

__device__ __forceinline__ float sigmoidf_(float v) {
    return 1.0f / (1.0f + __expf(-v));
}

__device__ __forceinline__ void wait_async(void) {
#if USE_ASYNC
#if __has_builtin(__builtin_amdgcn_s_wait_asynccnt)
    __builtin_amdgcn_s_wait_asynccnt(0);
#else
    asm volatile("s_wait_asynccnt 0x0" ::: "memory");
#endif
    asm volatile("" ::: "memory");
#endif
}
__device__ __forceinline__ void wait_async_le4(void) {
#if USE_ASYNC
#if __has_builtin(__builtin_amdgcn_s_wait_asynccnt)
    __builtin_amdgcn_s_wait_asynccnt(4);
#else
    asm volatile("s_wait_asynccnt 0x4" ::: "memory");
#endif
    asm volatile("" ::: "memory");
#endif
}

#if USE_ASYNC
// Copy one contiguous 2KB x tile (16 rows x 128B) into padded LDS rows.
// Lane i moves 16B: row = i/8, 16B-chunk (i%8) of that row. 4 async b128 ops.
__device__ __forceinline__ void tile_load_async(float* lbuf, const float* gsrc,
                                                int lane) {
    const int sub  = lane & 7;        // 16B chunk within a row
    const int row0 = lane >> 3;       // rows 0..3 for instruction 0
#pragma unroll
    for (int q = 0; q < 4; ++q) {
        int row = row0 + 4 * q;
        const float* g = gsrc + (size_t)(row * D_) + sub * 4;
        float*       l = lbuf + row * XROW + sub * 4;
        __builtin_amdgcn_global_load_async_to_lds_b128(
            (gv4i*)g, (lv4i*)l, 0, 0);
    }
}
#endif

// ---------------------------------------------------------------------------
// Kernel 1: per-(b,s) terms that depend only on prev_state (B*S = 1024 work).
// ---------------------------------------------------------------------------
__global__ void gtm_precompute(const float* __restrict__ prev_state,
                               const float* __restrict__ W_s,
                               const float* __restrict__ W_gs,
                               const float* __restrict__ b_gs,
                               const float* __restrict__ W_dc,
                               const float* __restrict__ b_dc,
                               float* __restrict__ ctx_const,
                               float* __restrict__ s_mix,
                               float* __restrict__ gate_state) {
    int tid = blockIdx.x * blockDim.x + threadIdx.x;
    if (tid >= B_ * S_) return;
    int b = tid / S_, s = tid % S_;
    const float* ps = prev_state + b * S_;
    float d1 = 0.f, d2 = 0.f, d3 = 0.f;
#pragma unroll
    for (int j = 0; j < S_; ++j) {
        float p = ps[j];
        d1 += p * W_dc[s * (D_ + S_) + D_ + j];   // decay_ctrl, state part
        d2 += p * W_s[s * S_ + j];                // s_proj
        d3 += p * W_gs[s * S_ + j];               // gate_s
    }
    ctx_const[tid]  = d1 + b_dc[s];
    s_mix[tid]      = d2 * ps[s];
    gate_state[tid] = sigmoidf_(d3 + b_gs[s]);
}

// ---------------------------------------------------------------------------
// Kernel 2: fused main pass. One wave per 16-token tile.
//   delta/ctx/gate: M=16 tokens, N=16 s, K=32 -> 8 V_WMMA_F32_16X16X4_F32 each
//   out:            M=16 tokens, N=16 d (x2),  K=16 -> 4 each
// x tiles arrive via double-buffered GLOBAL_LOAD_ASYNC_TO_LDS_B128 (ASYNCcnt),
// fully coalesced 512B per instruction; A-fragments read from padded LDS.
// ---------------------------------------------------------------------------
__launch_bounds__(256)
__global__ void gtm_main(const float* __restrict__ x,
                         const float* __restrict__ W_in,
                         const float* __restrict__ W_out,
                         const float* __restrict__ W_gx,
                         const float* __restrict__ b_gx,
                         const float* __restrict__ W_dc,
                         const float* __restrict__ ctx_const,
                         const float* __restrict__ s_mix,
                         const float* __restrict__ gate_state,
                         float* __restrict__ out,
                         float* __restrict__ h_last) {
#if USE_ASYNC
    __shared__ float xbuf[8][2][16 * XROW];   // per-wave double buffer (36.9KB)
#endif
    __shared__ float htile[8][16 * 18];       // per-wave h transpose tile

    const int lane    = threadIdx.x & 31;
    const int wid     = threadIdx.x >> 5;
    const int lane_lo = lane & 15;      // N index (s or d) / A row
    const int lane_hi = lane >> 4;      // which K-pair of the chunk
    const int koff    = 2 * lane_hi;

    const int gwave  = blockIdx.x * (blockDim.x >> 5) + wid;
    const int nwaves = gridDim.x * (blockDim.x >> 5);

    // ---- hoist weight B-fragments into registers (once per wave) ----
    v2f win_b[8], wdc_b[8], wgx_b[8];
#pragma unroll
    for (int c = 0; c < 8; ++c) {
        int d0 = 4 * c + koff;                         // K = d
        win_b[c] = *(const v2f*)(W_in + lane_lo * D_ + d0);
        wdc_b[c] = *(const v2f*)(W_dc + lane_lo * (D_ + S_) + d0);
        wgx_b[c] = *(const v2f*)(W_gx + lane_lo * D_ + d0);
    }
    v2f wout_b[2][4];
#pragma unroll
    for (int hf = 0; hf < 2; ++hf) {
#pragma unroll
        for (int c = 0; c < 4; ++c) {
            int dcol = hf * 16 + lane_lo;              // N = d
            int k0   = 4 * c + koff;                   // K = s
            wout_b[hf][c] = *(const v2f*)(W_out + dcol * S_ + k0);
        }
    }
    const float bgx_s = b_gx[lane_lo];

#if USE_ASYNC
    int parity = 0;
    if (gwave < NTILES) {
        int b0 = gwave >> 9, tt0 = (gwave & 511) << 4;
        tile_load_async(xbuf[wid][0], x + (size_t)(b0 * T_ + tt0) * D_, lane);
    }
#endif

    for (int tile = gwave; tile < NTILES; tile += nwaves) {
        const int b  = tile >> 9;            // / (T/16)
        const int t0 = (tile & 511) << 4;

        v2f a[8];
#if USE_ASYNC
        {
            int next = tile + nwaves;
            if (next < NTILES) {
                int nb = next >> 9, nt = (next & 511) << 4;
                tile_load_async(xbuf[wid][parity ^ 1],
                                x + (size_t)(nb * T_ + nt) * D_, lane);
                wait_async_le4();            // current tile's 4 copies done
            } else {
                wait_async();
            }
            const float* cur = xbuf[wid][parity];
#pragma unroll
            for (int c = 0; c < 8; ++c)
                a[c] = *(const v2f*)(cur + lane_lo * XROW + 4 * c + koff);
            parity ^= 1;
        }
#else
        {
            const float* xrow = x + ((size_t)(b * T_ + t0) + lane_lo) * D_;
#pragma unroll
            for (int c = 0; c < 8; ++c)
                a[c] = *(const v2f*)(xrow + 4 * c + koff);
        }
#endif

        v8f delta = {}; v8f ctx = {}; v8f gate = {};
#pragma unroll
        for (int c = 0; c < 8; ++c) {
            delta = __builtin_amdgcn_wmma_f32_16x16x4_f32(
                false, a[c], false, win_b[c], (short)0, delta, false, false);
            ctx = __builtin_amdgcn_wmma_f32_16x16x4_f32(
                false, a[c], false, wdc_b[c], (short)0, ctx, false, false);
            gate = __builtin_amdgcn_wmma_f32_16x16x4_f32(
                false, a[c], false, wgx_b[c], (short)0, gate, false, false);
        }

        // Per-batch terms: lane's column is s = lane_lo.
        const int   bs  = b * S_ + lane_lo;
        const float cc  = ctx_const[bs];
        const float sm  = s_mix[bs];
        const float gsv = gate_state[bs];

        // Elementwise epilogue + stage h in LDS for the C->A transpose.
        float* myl = htile[wid];
#pragma unroll
        for (int r = 0; r < 8; ++r) {
            int m = r + 8 * lane_hi;                 // token row within tile
            float dec = __expf(-BASE_DECAY * sigmoidf_(ctx[r] + cc));
            float h   = delta[r] + sm * dec;
            h *= sigmoidf_(gate[r] + bgx_s) * gsv;
            myl[m * 18 + lane_lo] = h;
            if (t0 + m == T_ - 1)
                h_last[bs] = h;                      // h[:, -1, :]
        }
        // Same-wave DS ops are in-order; private LDS slice -> no barrier.

        // Re-read h as A fragments (K = s, even K -> 8B aligned b64 loads).
        v2f ha[4];
#pragma unroll
        for (int c = 0; c < 4; ++c)
            ha[c] = *(const v2f*)(myl + lane_lo * 18 + 4 * c + koff);

        v8f o0 = {}; v8f o1 = {};
#pragma unroll
        for (int c = 0; c < 4; ++c) {
            o0 = __builtin_amdgcn_wmma_f32_16x16x4_f32(
                false, ha[c], false, wout_b[0][c], (short)0, o0, false, false);
            o1 = __builtin_amdgcn_wmma_f32_16x16x4_f32(
                false, ha[c], false, wout_b[1][c], (short)0, o1, false, false);
        }

        // Coalesced stores: lanes 0-15 / 16-31 each cover a contiguous 64B run.
        float* orow = out + (size_t)(b * T_ + t0) * D_;
#pragma unroll
        for (int r = 0; r < 8; ++r) {
            int m = r + 8 * lane_hi;
            orow[m * D_ + lane_lo]      = o0[r];
            orow[m * D_ + 16 + lane_lo] = o1[r];
        }
    }
}

// ---------------------------------------------------------------------------
extern "C" void kernel_launch(void* const* d_in, const int* in_sizes, int n_in,
                              void* d_out, int out_size, void* d_ws, size_t ws_size,
                              hipStream_t stream) {
    (void)in_sizes; (void)n_in; (void)out_size; (void)ws_size;
    const float* x    = (const float*)d_in[0];
    const float* prev = (const float*)d_in[1];
    const float* W_in = (const float*)d_in[2];
    const float* W_s  = (const float*)d_in[3];
    const float* W_out= (const float*)d_in[4];
    const float* W_gx = (const float*)d_in[5];
    const float* b_gx = (const float*)d_in[6];
    const float* W_gs = (const float*)d_in[7];
    const float* b_gs = (const float*)d_in[8];
    const float* W_dc = (const float*)d_in[9];
    const float* b_dc = (const float*)d_in[10];

    float* out    = (float*)d_out;
    float* h_last = out + (size_t)B_ * T_ * D_;   // tuple output #2

    float* ctx_const  = (float*)d_ws;
    float* s_mix      = ctx_const + B_ * S_;
    float* gate_state = s_mix + B_ * S_;

    gtm_precompute<<<1, B_ * S_, 0, stream>>>(prev, W_s, W_gs, b_gs, W_dc, b_dc,
                                              ctx_const, s_mix, gate_state);
    // 512 blocks x 8 waves = 4096 waves; exactly 8 tiles each (32768 tiles).
    gtm_main<<<512, 256, 0, stream>>>(x, W_in, W_out, W_gx, b_gx, W_dc,
                                      ctx_const, s_mix, gate_state, out, h_last);
}